// OhemCELoss_39548058862075
// MI455X (gfx1250) — compile-verified
//
#include <hip/hip_runtime.h>
#include <hip/hip_bf16.h>

// ---------------- problem constants (reference shapes are fixed) -------------
#define NCH        19
#define PLANE      (512 * 1024)            // H*W = 524288
#define NBATCH     8
#define N_PIX      (NBATCH * PLANE)        // 4194304
#define TILE       1024                    // pixels per TDM tile (divides PLANE)
#define TPB        256
#define NBLK_MAIN  512
#define TILES_TOTAL     (N_PIX / TILE)             // 4096
#define TILES_PER_BLK   (TILES_TOTAL / NBLK_MAIN)  // 8
#define TILES_PER_BATCH (PLANE / TILE)             // 512
#define SMEM_BYTES (2 * NCH * TILE * 4)    // 155648 B double buffer (dynamic LDS)
#define NBLK_GT    512
#define RANK_K     8193u                   // (N_MIN+1)-th largest -> vals[8192]
#define THRESH_F   0.35667494393873245f    // -log(0.7)
#define IGNORE_IDX (-100)

// ---------------- TDM (Tensor Data Mover) support ---------------------------
#if defined(__has_builtin)
#if __has_builtin(__builtin_amdgcn_tensor_load_to_lds)
#define HAVE_TDM 1
#endif
#endif
#ifndef HAVE_TDM
#define HAVE_TDM 0
#endif

#if HAVE_TDM
typedef unsigned int u32x4 __attribute__((ext_vector_type(4)));
typedef int          i32x4 __attribute__((ext_vector_type(4)));
typedef int          i32x8 __attribute__((ext_vector_type(8)));

// Issue a TDM load of a [NCH x TILE] tile: 19 rows of 1024 contiguous floats,
// row stride = PLANE floats, packed contiguously into LDS at ldsOff.
// D# packing per CDNA5 ISA.
__device__ __forceinline__ void tdm_load_tile(const float* gsrc, unsigned ldsOff) {
  unsigned long long ga = (unsigned long long)(uintptr_t)gsrc;
  u32x4 g0;
  g0.x = 1u;                                                 // count=1 (valid), user mode
  g0.y = ldsOff;                                             // lds_addr (bytes)
  g0.z = (unsigned)ga;                                       // global_addr[31:0]
  g0.w = (unsigned)((ga >> 32) & 0x01ffffffu) | (2u << 30);  // addr[56:32] | type=2
  const unsigned td0 = (unsigned)TILE;   // tensor_dim0 (no OOB: == tile_dim0)
  const unsigned td1 = (unsigned)NCH;    // tensor_dim1
  const unsigned long long st0 = (unsigned long long)PLANE;  // dim0 stride (elems)
  i32x8 g1;
  g1.s0 = (int)(2u << 16);                                   // wg_mask=0, data_size=2 (4B)
  g1.s1 = (int)((td0 & 0xffffu) << 16);                      // tensor_dim0[15:0] @ [63:48]
  g1.s2 = (int)((td0 >> 16) | ((td1 & 0xffffu) << 16));      // dim0[31:16] | dim1[15:0]
  g1.s3 = (int)((td1 >> 16) | (((unsigned)TILE & 0xffffu) << 16)); // dim1[31:16]|tile_dim0
  g1.s4 = (int)((unsigned)NCH & 0xffffu);                    // tile_dim1=19, tile_dim2=0
  g1.s5 = (int)(unsigned)(st0 & 0xffffffffu);                // stride0[31:0]
  g1.s6 = (int)(unsigned)((st0 >> 32) & 0xffffu);            // stride0[47:32], dim1_stride lo=0
  g1.s7 = 0;
  i32x4 gz = (i32x4)0;
#if defined(__clang_major__) && __clang_major__ >= 23
  i32x8 gz8 = (i32x8)0;
  __builtin_amdgcn_tensor_load_to_lds(g0, g1, gz, gz, gz8, 0);
#else
  __builtin_amdgcn_tensor_load_to_lds(g0, g1, gz, gz, 0);
#endif
}
#endif // HAVE_TDM

__device__ __forceinline__ const float* tile_gptr(const float* logit, int tile) {
  int b  = tile / TILES_PER_BATCH;
  int p0 = (tile % TILES_PER_BATCH) * TILE;
  return logit + (size_t)b * NCH * PLANE + p0;
}

// ---------------- K1: streaming CE loss (TDM double-buffered) ----------------
__global__ void __launch_bounds__(TPB) ohem_main_kernel(
    const float* __restrict__ logit, const int* __restrict__ label,
    float* __restrict__ loss, float4* __restrict__ blkPart)
{
  extern __shared__ float smem[];
  const int t = threadIdx.x;
  const int firstTile = blockIdx.x * TILES_PER_BLK;

  float pSum = 0.f, pValid = 0.f, pAbove = 0.f, pAcnt = 0.f;

#if HAVE_TDM
  if (t == 0) {
    tdm_load_tile(tile_gptr(logit, firstTile),
                  (unsigned)(uintptr_t)smem);
  }
#endif

  for (int i = 0; i < TILES_PER_BLK; ++i) {
    const int tile = firstTile + i;
    float* buf = smem + (size_t)(i & 1) * (NCH * TILE);

#if HAVE_TDM
    if (t == 0) {
      if (i + 1 < TILES_PER_BLK) {
        // prefetch next tile into the other buffer, then wait for current
        tdm_load_tile(tile_gptr(logit, tile + 1),
                      (unsigned)(uintptr_t)(smem + (size_t)((i + 1) & 1) * (NCH * TILE)));
        __builtin_amdgcn_s_wait_tensorcnt((short)1);
      } else {
        __builtin_amdgcn_s_wait_tensorcnt((short)0);
      }
    }
    __syncthreads();
#else
    { // cooperative fallback copy HBM -> LDS
      const float* g = tile_gptr(logit, tile);
      for (int idx = t; idx < NCH * (TILE / 4); idx += TPB) {
        int c = idx >> 8;          // idx / (TILE/4)
        int x = idx & 255;         // idx % (TILE/4)
        ((float4*)buf)[idx] = ((const float4*)(g + (size_t)c * PLANE))[x];
      }
      __syncthreads();
    }
#endif

    // each thread: 4 consecutive pixels
    const int4 lab4 = ((const int4*)(label + (size_t)tile * TILE))[t];
    const float4* lb4 = (const float4*)buf;

    float4 m  = make_float4(-3.4e38f, -3.4e38f, -3.4e38f, -3.4e38f);
    float4 xl = make_float4(0.f, 0.f, 0.f, 0.f);
#pragma unroll
    for (int c = 0; c < NCH; ++c) {
      float4 v = lb4[c * (TILE / 4) + t];
      m.x = fmaxf(m.x, v.x); m.y = fmaxf(m.y, v.y);
      m.z = fmaxf(m.z, v.z); m.w = fmaxf(m.w, v.w);
      if (lab4.x == c) xl.x = v.x;
      if (lab4.y == c) xl.y = v.y;
      if (lab4.z == c) xl.z = v.z;
      if (lab4.w == c) xl.w = v.w;
    }
    float4 s = make_float4(0.f, 0.f, 0.f, 0.f);
#pragma unroll
    for (int c = 0; c < NCH; ++c) {
      float4 v = lb4[c * (TILE / 4) + t];
      s.x += __expf(v.x - m.x); s.y += __expf(v.y - m.y);
      s.z += __expf(v.z - m.z); s.w += __expf(v.w - m.w);
    }
    float4 L;
    L.x = (lab4.x != IGNORE_IDX) ? (m.x + __logf(s.x) - xl.x) : 0.f;
    L.y = (lab4.y != IGNORE_IDX) ? (m.y + __logf(s.y) - xl.y) : 0.f;
    L.z = (lab4.z != IGNORE_IDX) ? (m.z + __logf(s.z) - xl.z) : 0.f;
    L.w = (lab4.w != IGNORE_IDX) ? (m.w + __logf(s.w) - xl.w) : 0.f;

    ((float4*)(loss + (size_t)tile * TILE))[t] = L;

    pSum   += L.x + L.y + L.z + L.w;
    pValid += (float)((lab4.x != IGNORE_IDX) + (lab4.y != IGNORE_IDX) +
                      (lab4.z != IGNORE_IDX) + (lab4.w != IGNORE_IDX));
    if (L.x > THRESH_F) { pAbove += L.x; pAcnt += 1.f; }
    if (L.y > THRESH_F) { pAbove += L.y; pAcnt += 1.f; }
    if (L.z > THRESH_F) { pAbove += L.z; pAcnt += 1.f; }
    if (L.w > THRESH_F) { pAbove += L.w; pAcnt += 1.f; }

    __syncthreads();  // buffer reuse guard (next TDM overwrites this buffer)
  }

  // deterministic fixed-tree block reduction (reuse LDS)
  float4* red = (float4*)smem;
  red[t] = make_float4(pSum, pValid, pAbove, pAcnt);
  __syncthreads();
  for (int sft = TPB / 2; sft > 0; sft >>= 1) {
    if (t < sft) {
      float4 o = red[t + sft];
      float4 a = red[t];
      a.x += o.x; a.y += o.y; a.z += o.z; a.w += o.w;
      red[t] = a;
    }
    __syncthreads();
  }
  if (t == 0) blkPart[blockIdx.x] = red[0];
}

// ---------------- K0: init workspace state ----------------------------------
__global__ void init_kernel(unsigned* __restrict__ hist, unsigned* __restrict__ state) {
  hist[threadIdx.x] = 0u;
  if (threadIdx.x == 0) { state[0] = 0u; state[1] = RANK_K; }
}

// ---------------- radix-select: histogram stage ------------------------------
__global__ void __launch_bounds__(256) hist_kernel(
    const float* __restrict__ loss, unsigned* __restrict__ hist,
    const unsigned* __restrict__ state, int byteIdx, int n)
{
  __shared__ unsigned h[256];
  h[threadIdx.x] = 0u;
  __syncthreads();
  const unsigned prefix = state[0];
  const int stride = gridDim.x * blockDim.x;
  for (int i = blockIdx.x * blockDim.x + threadIdx.x; i < n; i += stride) {
    unsigned u = __float_as_uint(loss[i]);  // loss >= 0 -> bits monotonic
    bool match = (byteIdx == 3) || ((u >> ((byteIdx + 1) * 8)) == prefix);
    if (match) atomicAdd(&h[(u >> (byteIdx * 8)) & 255u], 1u);
  }
  __syncthreads();
  atomicAdd(&hist[threadIdx.x], h[threadIdx.x]);
}

// ---------------- radix-select: serial scan stage ----------------------------
__global__ void scan_kernel(unsigned* __restrict__ hist, unsigned* __restrict__ state,
                            float* __restrict__ kth, int byteIdx)
{
  unsigned r = state[1];
  unsigned prefix = state[0];
  unsigned cum = 0, sel = 0;
  for (int b = 255; b >= 0; --b) {
    unsigned cb = hist[b];
    if (cum + cb >= r) { sel = (unsigned)b; r -= cum; break; }
    cum += cb;
  }
  prefix = (prefix << 8) | sel;
  state[0] = prefix;
  state[1] = r;
  if (byteIdx == 0) kth[0] = __uint_as_float(prefix);  // exact 8193-rd largest
  for (int b = 0; b < 256; ++b) hist[b] = 0u;          // ready for next stage
}

// ---------------- pass: sum/count of loss > kth ------------------------------
__global__ void __launch_bounds__(256) gt_kernel(
    const float* __restrict__ loss, const float* __restrict__ kth_p,
    float2* __restrict__ blk2, int n)
{
  const float k = kth_p[0];
  float s = 0.f, c = 0.f;
  const int stride = gridDim.x * blockDim.x;
  for (int i = blockIdx.x * blockDim.x + threadIdx.x; i < n; i += stride) {
    float v = loss[i];
    if (v > k) { s += v; c += 1.f; }
  }
  __shared__ float2 red[256];
  red[threadIdx.x] = make_float2(s, c);
  __syncthreads();
  for (int sft = 128; sft > 0; sft >>= 1) {
    if (threadIdx.x < sft) {
      red[threadIdx.x].x += red[threadIdx.x + sft].x;
      red[threadIdx.x].y += red[threadIdx.x + sft].y;
    }
    __syncthreads();
  }
  if (threadIdx.x == 0) blk2[blockIdx.x] = red[0];
}

// ---------------- finalize ---------------------------------------------------
__global__ void __launch_bounds__(256) finalize_kernel(
    const float4* __restrict__ blkPart, const float2* __restrict__ blk2,
    const float* __restrict__ kth_p, float* __restrict__ out)
{
  __shared__ float4 r4[256];
  __shared__ float2 r2[256];
  const int t = threadIdx.x;
  float4 a = make_float4(0.f, 0.f, 0.f, 0.f);
  for (int i = t; i < NBLK_MAIN; i += 256) {
    float4 v = blkPart[i];
    a.x += v.x; a.y += v.y; a.z += v.z; a.w += v.w;
  }
  float2 b = make_float2(0.f, 0.f);
  for (int i = t; i < NBLK_GT; i += 256) {
    float2 v = blk2[i];
    b.x += v.x; b.y += v.y;
  }
  r4[t] = a; r2[t] = b;
  __syncthreads();
  for (int sft = 128; sft > 0; sft >>= 1) {
    if (t < sft) {
      r4[t].x += r4[t + sft].x; r4[t].y += r4[t + sft].y;
      r4[t].z += r4[t + sft].z; r4[t].w += r4[t + sft].w;
      r2[t].x += r2[t + sft].x; r2[t].y += r2[t + sft].y;
    }
    __syncthreads();
  }
  if (t == 0) {
    const float sum_loss = r4[0].x, cnt_valid = r4[0].y;
    const float sum_above = r4[0].z, cnt_above = r4[0].w;
    const float sum_gt = r2[0].x, cnt_gt = r2[0].y;
    const float kth = kth_p[0];
    const float normal = sum_loss / fmaxf(cnt_valid, 1.f);
    const float above_mean = sum_above / fmaxf(cnt_above, 1.f);
    // top-8192 = all values > kth plus (8192 - cnt_gt) copies of kth (ties)
    const float topk_mean = (sum_gt + (8192.f - cnt_gt) * kth) * (1.f / 8192.f);
    out[0] = normal;
    out[1] = (kth > THRESH_F) ? above_mean : topk_mean;
  }
}

// ---------------- host-side launch -------------------------------------------
extern "C" void kernel_launch(void* const* d_in, const int* in_sizes, int n_in,
                              void* d_out, int out_size, void* d_ws, size_t ws_size,
                              hipStream_t stream)
{
  const float* logit = (const float*)d_in[0];
  const int*   label = (const int*)d_in[1];
  float*       out   = (float*)d_out;
  const int n = N_PIX;

  // workspace layout
  float*    loss    = (float*)d_ws;                                   // 16 MB
  float4*   blkPart = (float4*)((char*)d_ws + (size_t)n * sizeof(float));
  float2*   blk2    = (float2*)(blkPart + NBLK_MAIN);
  unsigned* hist    = (unsigned*)(blk2 + NBLK_GT);
  unsigned* state   = hist + 256;
  float*    kth     = (float*)(state + 2);

  init_kernel<<<1, 256, 0, stream>>>(hist, state);
  ohem_main_kernel<<<NBLK_MAIN, TPB, SMEM_BYTES, stream>>>(logit, label, loss, blkPart);
  for (int b = 3; b >= 0; --b) {
    hist_kernel<<<1024, 256, 0, stream>>>(loss, hist, state, b, n);
    scan_kernel<<<1, 1, 0, stream>>>(hist, state, kth, b);
  }
  gt_kernel<<<NBLK_GT, 256, 0, stream>>>(loss, kth, blk2, n);
  finalize_kernel<<<1, 256, 0, stream>>>(blkPart, blk2, kth, out);
}